// Deconv2DVariableWeights_84628035600827
// MI455X (gfx1250) — compile-verified
//
#include <hip/hip_runtime.h>
#include <math.h>

// ---------------------------------------------------------------------------
// Deconv2DVariableWeights on MI455X (gfx1250), fp32 WMMA path.
//   B=32, C=128, H=W=32, ZDIM=256, NB=C*C*9=147456, K_conv=C*9=1152
// Stage 1: h = relu(z @ dense_w + dense_b)      (WMMA f32 16x16x4, M=32)
// Stage 2: batch-norm over axis 0 + gamma/beta  (in-place, streaming)
// Stage 3: per-sample implicit-GEMM 3x3 conv    (WMMA f32 16x16x4,
//          async global->LDS staging, s-major K order: zero per-step VALU)
// ---------------------------------------------------------------------------

typedef __attribute__((ext_vector_type(2))) float v2f;
typedef __attribute__((ext_vector_type(8))) float v8f;

#define BB     32
#define CC     128
#define HW     1024      // 32*32
#define ZD     256
#define NB     147456    // 128*128*9
#define KC     1152      // 128*9
#define BN_EPS 1e-6f

// LDS staging geometry: 4 image rows x 128 channels x 32 cols, row stride 36
// floats (144B, 16B-aligned for b128 async stores), 4 zero pad cols per row
// and a 16B zero prefix so iw in [-1,32] needs no bounds check.
#define SROW   36
#define SSTEP  (4 * SROW)                // floats per channel slab
#define SBASE  4
#define SELEMS (SBASE + CC * SSTEP)      // 18436 floats = 73744 bytes

// ---------------------------------------------------------------------------
// Kernel 1: hyper-network GEMM + bias + ReLU (memory-bound on 151MB W stream).
// ---------------------------------------------------------------------------
__global__ __launch_bounds__(256) void hyper_gemm_relu(
    const float* __restrict__ z, const float* __restrict__ Wd,
    const float* __restrict__ bd, float* __restrict__ h)
{
  const int lane = threadIdx.x & 31;
  const int wave = threadIdx.x >> 5;
  const int col  = lane & 15;
  const int hi   = lane >> 4;
  const int n0   = (blockIdx.x * 8 + wave) * 16;
  const int n    = n0 + col;

  v8f c0 = {};   // rows 0..15
  v8f c1 = {};   // rows 16..31

  for (int kk = 0; kk < ZD; kk += 4) {
    const int ka = kk + 2 * hi;
    v2f a0, a1, b;
    a0.x = z[col * ZD + ka];
    a0.y = z[col * ZD + ka + 1];
    a1.x = z[(16 + col) * ZD + ka];
    a1.y = z[(16 + col) * ZD + ka + 1];
    b.x  = Wd[(size_t)ka       * NB + n];
    b.y  = Wd[(size_t)(ka + 1) * NB + n];
    if (kk + 8 < ZD)
      __builtin_prefetch(&Wd[(size_t)(kk + 8 + 2 * hi) * NB + n], 0, 1);
    c0 = __builtin_amdgcn_wmma_f32_16x16x4_f32(false, a0, false, b,
                                               (short)0, c0, false, false);
    c1 = __builtin_amdgcn_wmma_f32_16x16x4_f32(false, a1, false, b,
                                               (short)0, c1, false, false);
  }

  const float bias = bd[n];
#pragma unroll
  for (int r = 0; r < 8; ++r) {
    const int m = r + 8 * hi;
    float v0 = c0[r] + bias;
    float v1 = c1[r] + bias;
    h[(size_t)m        * NB + n] = v0 > 0.f ? v0 : 0.f;
    h[(size_t)(16 + m) * NB + n] = v1 > 0.f ? v1 : 0.f;
  }
}

// ---------------------------------------------------------------------------
// Kernel 2: batch-norm over the batch axis + gamma/beta, in place on h.
// ---------------------------------------------------------------------------
__global__ __launch_bounds__(256) void bn_affine_inplace(
    float* __restrict__ h, const float* __restrict__ gamma,
    const float* __restrict__ beta)
{
  const int j = blockIdx.x * 256 + threadIdx.x;
  if (j >= NB) return;

  float v[BB];
  float s = 0.f;
#pragma unroll
  for (int m = 0; m < BB; ++m) { v[m] = h[(size_t)m * NB + j]; s += v[m]; }
  const float mean = s * (1.f / BB);
  float q = 0.f;
#pragma unroll
  for (int m = 0; m < BB; ++m) { const float d = v[m] - mean; q += d * d; }
  const float var = q * (1.f / BB);
  const float inv = 1.f / (sqrtf(var) + BN_EPS);
  const float g = gamma[j], bt = beta[j];
#pragma unroll
  for (int m = 0; m < BB; ++m)
    h[(size_t)m * NB + j] = g * ((v[m] - mean) * inv) + bt;
}

// ---------------------------------------------------------------------------
// Kernel 3: per-sample conv as implicit GEMM + residual + channel bias.
// K-dimension iterated s-major (4 channels per WMMA) with the 9 filter taps
// fully unrolled at compile time -> all tap/pixel/s+1 variation becomes load
// immediates; per outer step only two pointer bumps of VALU work.
// ---------------------------------------------------------------------------
__global__ __launch_bounds__(256) void hyperconv_wmma(
    const float* __restrict__ x, const float* __restrict__ wn,
    const float* __restrict__ cbias, float* __restrict__ out)
{
  extern __shared__ float smem[];

  const int lane = threadIdx.x & 31;
  const int wave = threadIdx.x >> 5;
  const int col  = lane & 15;
  const int hi   = lane >> 4;

  const int b   = blockIdx.y;          // sample
  const int f0  = wave * 16;           // output-channel tile
  const int nbp = blockIdx.x * 64;     // 64 consecutive pixels (2 image rows)
  const int p0  = nbp >> 5;            // first image row of this group

  const float* wb = wn + (size_t)b * NB;        // [s][f][3][3]
  const float* xb = x  + (size_t)b * CC * HW;   // [s][32][32]

  // ---- Stage x slab into LDS --------------------------------------------
  for (int i = threadIdx.x; i < SELEMS; i += 256) smem[i] = 0.f;
  __syncthreads();
#pragma unroll
  for (int i = 0; i < 16; ++i) {
    const int idx = i * 256 + threadIdx.x;   // 0..4095 chunk id
    const int s   = idx >> 5;                // channel
    const int rem = idx & 31;
    const int r   = rem >> 3;                // slab row 0..3
    const int c4  = rem & 7;                 // 16B chunk within row
    const int ih  = p0 - 1 + r;
    if (ih >= 0 && ih < 32) {
      unsigned lds_off =
          (unsigned)(uintptr_t)(smem + SBASE + s * SSTEP + r * SROW + c4 * 4);
      unsigned long long gaddr =
          (unsigned long long)(uintptr_t)(xb + ((s * 32 + ih) * 32 + c4 * 4));
      asm volatile("global_load_async_to_lds_b128 %0, %1, off"
                   :: "v"(lds_off), "v"(gaddr) : "memory");
    }
  }
  asm volatile("s_wait_asynccnt 0" ::: "memory");
  __syncthreads();

  // ---- Implicit GEMM: D(16f x 64pix) += A(16f x K) * B(K x 64pix) -------
  v8f acc[4] = {v8f{}, v8f{}, v8f{}, v8f{}};
  const int f = f0 + col;

  // loop-invariant per-tile pixel offsets inside the LDS slab
  int qoff[4];
#pragma unroll
  for (int t = 0; t < 4; ++t)
    qoff[t] = (t >> 1) * SROW + col + 16 * (t & 1);

  // per-lane running bases: A at channel (2*hi), B at channel (2*hi)
  const float* aptr = wb + (size_t)(2 * hi) * KC + f * 9;
  const float* bptr = smem + SBASE + (2 * hi) * SSTEP;

  for (int sb = 0; sb < CC; sb += 4) {
#pragma unroll
    for (int w = 0; w < 9; ++w) {            // tap (u,v) compile-time
      const int u  = w / 3;
      const int vv = w - u * 3;
      v2f a;                                  // K rows 2hi, 2hi+1 of 4-slice
      a.x = aptr[w];                          //   s = sb + 2*hi
      a.y = aptr[KC + w];                     //   s = sb + 2*hi + 1
      const int bo = u * SROW + vv - 1;       // tap offset (halo pre-zeroed)
#pragma unroll
      for (int t = 0; t < 4; ++t) {
        v2f bm;
        bm.x = bptr[bo + qoff[t]];
        bm.y = bptr[SSTEP + bo + qoff[t]];
        acc[t] = __builtin_amdgcn_wmma_f32_16x16x4_f32(false, a, false, bm,
                                                       (short)0, acc[t],
                                                       false, false);
      }
    }
    aptr += 4 * KC;      // next 4 channels
    bptr += 4 * SSTEP;
  }

#pragma unroll
  for (int t = 0; t < 4; ++t) {
    const int n = nbp + t * 16 + col;
#pragma unroll
    for (int r = 0; r < 8; ++r) {
      const int fr = f0 + r + 8 * hi;
      const size_t oidx = (size_t)(b * CC + fr) * HW + n;
      out[oidx] = acc[t][r] + x[oidx] + cbias[fr];   // residual + bias
    }
  }
}

// ---------------------------------------------------------------------------
// Host-side launch. Inputs (setup_inputs order):
//   0:x (B,C,H,W) 1:z (B,ZDIM) 2:dense_w (ZDIM,NB) 3:dense_b (NB)
//   4:gamma (NB)  5:beta (NB)  6:b (C)
// Workspace: h = 32*NB floats (~18.9 MB), normalized in place.
// ---------------------------------------------------------------------------
extern "C" void kernel_launch(void* const* d_in, const int* in_sizes, int n_in,
                              void* d_out, int out_size, void* d_ws, size_t ws_size,
                              hipStream_t stream) {
  (void)in_sizes; (void)n_in; (void)out_size; (void)ws_size;
  const float* x     = (const float*)d_in[0];
  const float* z     = (const float*)d_in[1];
  const float* dw    = (const float*)d_in[2];
  const float* db    = (const float*)d_in[3];
  const float* gamma = (const float*)d_in[4];
  const float* beta  = (const float*)d_in[5];
  const float* cbias = (const float*)d_in[6];
  float* out = (float*)d_out;
  float* h   = (float*)d_ws;                 // 32*NB floats

  hyper_gemm_relu<<<NB / (16 * 8), 256, 0, stream>>>(z, dw, db, h);
  bn_affine_inplace<<<(NB + 255) / 256, 256, 0, stream>>>(h, gamma, beta);
  dim3 grid(HW / 64, BB);
  hyperconv_wmma<<<grid, 256, SELEMS * sizeof(float), stream>>>(x, h, cbias, out);
}